// SAGE_9483287789791
// MI455X (gfx1250) — compile-verified
//
#include <hip/hip_runtime.h>
#include <hip/hip_bf16.h>
#include <cstdint>
#include <cstddef>

// ---------------- problem constants (match reference) ----------------
#define N_NODES 50000
#define N_EDGES 625000
#define IN_C    128
#define HID_C   128
#define OUT_C   64

typedef float v2f __attribute__((ext_vector_type(2)));
typedef float v8f __attribute__((ext_vector_type(8)));

// ---------------- CSR build ----------------
__global__ void zero_i4_kernel(int4* __restrict__ p, int n4) {
    int t = blockIdx.x * blockDim.x + threadIdx.x;
    if (t < n4) p[t] = make_int4(0, 0, 0, 0);
}

__global__ void deg_count_kernel(const int* __restrict__ dst,
                                 int* __restrict__ deg, int nE) {
    int t = blockIdx.x * blockDim.x + threadIdx.x;
    if (t < nE) atomicAdd(&deg[dst[t]], 1);
}

// block-wise exclusive scan (256 elements per block), emits block totals
__global__ void scan_block_kernel(const int* __restrict__ deg,
                                  int* __restrict__ rowstart,
                                  int* __restrict__ bsums, int n) {
    __shared__ int sh[256];
    const int tid = threadIdx.x;
    const int i = blockIdx.x * 256 + tid;
    int v = (i < n) ? deg[i] : 0;
    sh[tid] = v;
    __syncthreads();
    for (int off = 1; off < 256; off <<= 1) {
        int t = (tid >= off) ? sh[tid - off] : 0;
        __syncthreads();
        sh[tid] += t;
        __syncthreads();
    }
    if (i < n) rowstart[i] = sh[tid] - v;          // exclusive
    if (tid == 255) bsums[blockIdx.x] = sh[255];   // block total
}

// single-block exclusive scan of block totals (nb <= 256)
__global__ void scan_sums_kernel(int* __restrict__ bsums, int nb) {
    __shared__ int sh[256];
    const int tid = threadIdx.x;
    int v = (tid < nb) ? bsums[tid] : 0;
    sh[tid] = v;
    __syncthreads();
    for (int off = 1; off < 256; off <<= 1) {
        int t = (tid >= off) ? sh[tid - off] : 0;
        __syncthreads();
        sh[tid] += t;
        __syncthreads();
    }
    if (tid < nb) bsums[tid] = sh[tid] - v;        // exclusive
}

// add block offsets; also initialize fill cursor
__global__ void scan_add_kernel(int* __restrict__ rowstart,
                                const int* __restrict__ bsums,
                                int* __restrict__ cursor, int n) {
    int i = blockIdx.x * blockDim.x + threadIdx.x;
    if (i < n) {
        int r = rowstart[i] + bsums[i >> 8];
        rowstart[i] = r;
        cursor[i] = r;
    }
}

__global__ void fill_adj_kernel(const int* __restrict__ src,
                                const int* __restrict__ dst,
                                int* __restrict__ cursor,
                                int* __restrict__ adj, int nE) {
    int t = blockIdx.x * blockDim.x + threadIdx.x;
    if (t < nE) {
        int pos = atomicAdd(&cursor[dst[t]], 1);
        adj[pos] = src[t];
    }
}

// ---------------- atomic-free gather + mean: one wave per destination node --
__global__ void sage_gather_kernel(const float* __restrict__ xin,  // [N,128]
                                   const int* __restrict__ rowstart,
                                   const int* __restrict__ deg,
                                   const int* __restrict__ adj,
                                   float* __restrict__ agg) {       // [N,128] mean
    const int t = blockIdx.x * blockDim.x + threadIdx.x;
    const int v = t >> 5;
    if (v >= N_NODES) return;
    const int lane = t & 31;

    const int start = rowstart[v];
    const int d = deg[v];

    float4 acc = make_float4(0.f, 0.f, 0.f, 0.f);
    for (int ibase = 0; ibase < d; ibase += 32) {
        int a = 0;
        if (ibase + lane < d) a = adj[start + ibase + lane];
        const int lim = min(32, d - ibase);
        for (int j = 0; j < lim; ++j) {
            const int s = __shfl(a, j, 32);
            const float4 f = *(const float4*)(xin + (size_t)s * 128 + lane * 4);
            acc.x += f.x; acc.y += f.y; acc.z += f.z; acc.w += f.w;
        }
    }
    const float inv = 1.0f / fmaxf((float)d, 1.0f);
    acc.x *= inv; acc.y *= inv; acc.z *= inv; acc.w *= inv;
    *(float4*)(agg + (size_t)v * 128 + lane * 4) = acc;
}

// ---------------- fused SAGE combine:  out = act( [agg_mean | x] @ [Wl;Wr] + b ) ----
// Persistent blocks: W=[Wl;Wr] staged in LDS ONCE per block, then grid-stride
// over node blocks. 256 threads = 8 wave32; each wave owns one 16x16 output
// tile, accumulated with V_WMMA_F32_16X16X4_F32 over K=256 (64 steps).
template <int OUT, bool RELU>
__global__ __launch_bounds__(256) void sage_combine_kernel(
    const float* __restrict__ xin,   // [N,128]
    const float* __restrict__ agg,   // [N,128] (already mean)
    const float* __restrict__ Wl,    // [128,OUT] row-major
    const float* __restrict__ Wr,    // [128,OUT] row-major
    const float* __restrict__ bias,  // [OUT]
    float* __restrict__ out)         // [N,OUT]
{
    constexpr int NT   = OUT / 16;   // N tiles per node block
    constexpr int MT   = 8 / NT;     // M tiles per block (8 waves total)
    constexpr int MBLK = MT * 16;    // nodes per block

    extern __shared__ float smem[];
    float* Ash = smem;               // [MBLK][256]  fused features
    float* Wsh = smem + MBLK * 256;  // [256][OUT]   [Wl ; Wr]

    const int tid = threadIdx.x;

    // stage W = [Wl ; Wr] (both contiguous row-major) as float4 — once
    {
        const float4* wl4 = (const float4*)Wl;
        const float4* wr4 = (const float4*)Wr;
        float4* w4 = (float4*)Wsh;
        constexpr int WQ = 128 * OUT / 4;
        for (int i = tid; i < WQ; i += 256) w4[i] = wl4[i];
        for (int i = tid; i < WQ; i += 256) w4[WQ + i] = wr4[i];
    }

    const int wave  = tid >> 5;
    const int lane  = tid & 31;
    const int half  = lane >> 4;   // 0: lanes 0-15 (K+0,1), 1: lanes 16-31 (K+2,3)
    const int l15   = lane & 15;
    const int mtile = wave / NT;
    const int ntile = wave % NT;

    const float* Arow = Ash + (mtile * 16 + l15) * 256;
    const float* Wcol = Wsh + ntile * 16 + l15;
    const int    col  = ntile * 16 + l15;
    const float  bb   = bias[col];

    const int stride = gridDim.x * MBLK;
    for (int base = blockIdx.x * MBLK; base < N_NODES; base += stride) {
        __syncthreads();   // previous iteration's readers done before refill
        // stage A rows: k<128 -> agg (mean), k>=128 -> x
        {
            float4* a4 = (float4*)Ash;
            constexpr int AQ = MBLK * 64;        // float4 count
            for (int i = tid; i < AQ; i += 256) {
                int m  = i >> 6;
                int kq = i & 63;
                int node = base + m;
                if (node >= N_NODES) node = N_NODES - 1;  // clamp (stores guarded)
                float4 v;
                if (kq < 32) v = ((const float4*)(agg + (size_t)node * 128))[kq];
                else         v = ((const float4*)(xin + (size_t)node * 128))[kq - 32];
                a4[i] = v;
            }
        }
        __syncthreads();

        v8f c = {};                  // f32 accumulator tile (8 VGPRs)
#pragma unroll 8
        for (int k0 = 0; k0 < 256; k0 += 4) {
            const int k = k0 + 2 * half;
            v2f a = *(const v2f*)(Arow + k);       // A[m][k], A[m][k+1]
            v2f b;
            b.x = Wcol[(size_t)k * OUT];           // W[k][n]
            b.y = Wcol[(size_t)(k + 1) * OUT];     // W[k+1][n]
            c = __builtin_amdgcn_wmma_f32_16x16x4_f32(
                    /*neg_a=*/false, a, /*neg_b=*/false, b,
                    /*c_mod=*/(short)0, c, /*reuse_a=*/false, /*reuse_b=*/false);
        }

#pragma unroll
        for (int v = 0; v < 8; ++v) {
            int row  = mtile * 16 + v + 8 * half;  // C/D layout: VGPR v -> M=v (+8 hi lanes)
            int node = base + row;
            float val = c[v] + bb;
            if (RELU) val = fmaxf(val, 0.0f);
            if (node < N_NODES) out[(size_t)node * OUT + col] = val;
        }
    }
}

// ---------------- launcher ----------------
extern "C" void kernel_launch(void* const* d_in, const int* in_sizes, int n_in,
                              void* d_out, int out_size, void* d_ws, size_t ws_size,
                              hipStream_t stream) {
    (void)in_sizes; (void)n_in; (void)out_size; (void)ws_size;

    const float* x   = (const float*)d_in[0];
    const int*   ei  = (const int*)d_in[1];
    const int*   src = ei;                 // edge_index[0]
    const int*   dst = ei + N_EDGES;       // edge_index[1]
    const float* Wl1 = (const float*)d_in[2];
    const float* Wr1 = (const float*)d_in[3];
    const float* b1  = (const float*)d_in[4];
    const float* Wl2 = (const float*)d_in[5];
    const float* Wr2 = (const float*)d_in[6];
    const float* b2  = (const float*)d_in[7];
    const float* Wl3 = (const float*)d_in[8];
    const float* Wr3 = (const float*)d_in[9];
    const float* b3  = (const float*)d_in[10];
    float* out = (float*)d_out;

    // ---- workspace layout ----
    char* ws = (char*)d_ws;
    int*  deg      = (int*)ws;                 // 50048 (padded)
    int*  rowstart = deg + 50048;
    int*  cursor   = rowstart + 50048;
    int*  bsums    = cursor + 50048;           // 256
    int*  adj      = bsums + 256;              // N_EDGES
    float* agg = (float*)(ws + ((size_t)4 << 20));          // [N,128]
    float* h1  = agg + (size_t)N_NODES * 128;               // [N,128]
    float* h2  = h1  + (size_t)N_NODES * 128;               // [N,128]

    const int NBLK   = (N_NODES + 255) / 256;      // 196 scan blocks
    const int G_E    = (N_EDGES + 255) / 256;      // edge-parallel grids
    const int G_GATH = (N_NODES * 32 + 255) / 256; // 6250

    // ---- CSR build (dst identical for all 3 layers) ----
    zero_i4_kernel<<<(50048 / 4 + 255) / 256, 256, 0, stream>>>((int4*)deg, 50048 / 4);
    deg_count_kernel<<<G_E, 256, 0, stream>>>(dst, deg, N_EDGES);
    scan_block_kernel<<<NBLK, 256, 0, stream>>>(deg, rowstart, bsums, N_NODES);
    scan_sums_kernel<<<1, 256, 0, stream>>>(bsums, NBLK);
    scan_add_kernel<<<NBLK, 256, 0, stream>>>(rowstart, bsums, cursor, N_NODES);
    fill_adj_kernel<<<G_E, 256, 0, stream>>>(src, dst, cursor, adj, N_EDGES);

    const size_t SH12 = (size_t)(16 * 256 + 256 * 128) * 4; // 144 KB
    const size_t SH3  = (size_t)(32 * 256 + 256 * 64) * 4;  //  96 KB
    const int GB12 = 640;   // persistent blocks; ~5 node-blocks each (3125 tiles)
    const int GB3  = 320;   // persistent blocks; ~5 node-blocks each (1563 tiles)

    // layer 1: x -> h1 (relu)
    sage_gather_kernel<<<G_GATH, 256, 0, stream>>>(x, rowstart, deg, adj, agg);
    sage_combine_kernel<128, true><<<GB12, 256, SH12, stream>>>(x, agg, Wl1, Wr1, b1, h1);

    // layer 2: h1 -> h2 (relu)
    sage_gather_kernel<<<G_GATH, 256, 0, stream>>>(h1, rowstart, deg, adj, agg);
    sage_combine_kernel<128, true><<<GB12, 256, SH12, stream>>>(h1, agg, Wl2, Wr2, b2, h2);

    // layer 3: h2 -> out (no relu)
    sage_gather_kernel<<<G_GATH, 256, 0, stream>>>(h2, rowstart, deg, adj, agg);
    sage_combine_kernel<64, false><<<GB3, 256, SH3, stream>>>(h2, agg, Wl3, Wr3, b3, out);
}